// OrthoConv_15779709845910
// MI455X (gfx1250) — compile-verified
//
#include <hip/hip_runtime.h>
#include <hip/hip_bf16.h>

// Problem constants (match reference)
constexpr int   N_NODES = 100000;
constexpr int   N_EDGES = 1600000;
constexpr int   IN_C    = 256;
constexpr int   OUT_C   = 128;
constexpr int   HALF_C  = OUT_C / 2;

typedef __attribute__((ext_vector_type(2))) float v2f;
typedef __attribute__((ext_vector_type(8))) float v8f;

// ---------------------------------------------------------------------------
// Zero-fill (graph-capture safe alternative to memset)
// ---------------------------------------------------------------------------
__global__ void zero_f32_kernel(float* __restrict__ p, long n) {
    long i = (long)blockIdx.x * blockDim.x + threadIdx.x;
    long stride = (long)gridDim.x * blockDim.x;
    for (; i < n; i += stride) p[i] = 0.0f;
}

// ---------------------------------------------------------------------------
// In-degree of target nodes (col): deg[col[e]] += 1
// ---------------------------------------------------------------------------
__global__ void degree_kernel(const int* __restrict__ col, float* __restrict__ deg) {
    int e = blockIdx.x * blockDim.x + threadIdx.x;
    if (e >= N_EDGES) return;
    atomicAdd(&deg[col[e]], 1.0f);
}

// ---------------------------------------------------------------------------
// Per-edge GCN normalization: ew = dinv(row) * dinv(col)
// ---------------------------------------------------------------------------
__global__ void edge_weight_kernel(const int* __restrict__ row,
                                   const int* __restrict__ col,
                                   const float* __restrict__ deg,
                                   float* __restrict__ ew) {
    int e = blockIdx.x * blockDim.x + threadIdx.x;
    if (e >= N_EDGES) return;
    float dr = deg[row[e]];
    float dc = deg[col[e]];
    float ir = dr > 0.0f ? rsqrtf(dr) : 0.0f;
    float ic = dc > 0.0f ? rsqrtf(dc) : 0.0f;
    ew[e] = ir * ic;
}

// ---------------------------------------------------------------------------
// Fold the negate-second-half + roll-by-half postprocess into the weight:
//   out_final[:, i] = (i < 64) ? -out[:, i+64] : out[:, i-64]
// so  W2[i, :] = (i < 64) ? -W[i+64, :] : W[i-64, :]
// ---------------------------------------------------------------------------
__global__ void fold_weight_kernel(const float* __restrict__ w,
                                   float* __restrict__ w2) {
    int idx = blockIdx.x * blockDim.x + threadIdx.x;
    if (idx >= OUT_C * IN_C) return;
    int i = idx / IN_C;
    int k = idx - i * IN_C;
    int j = (i < HALF_C) ? (i + HALF_C) : (i - HALF_C);
    float s = (i < HALF_C) ? -1.0f : 1.0f;
    w2[idx] = s * w[j * IN_C + k];
}

// ---------------------------------------------------------------------------
// WMMA GEMM: h[N, 128] = x[N, 256] @ W2[128, 256]^T   (fp32, 16x16x4 WMMA)
//
// Block = (32, 8): 8 wave32s, each wave computes one 16x16 output tile.
// 100000 rows / 16 = 6250 M-tiles exactly; 128/16 = 8 N-tiles.
//
// f32 A-fragment layout (ISA 7.12.2, 16x4):
//   lanes 0-15 : {A[M=lane][K+0], A[M=lane][K+1]}
//   lanes 16-31: {A[M=lane-16][K+2], A[M=lane-16][K+3]}
// B-fragment (4x16, K x N) mirrors this with lane -> N:
//   lanes 0-15 : {B[K+0][n], B[K+1][n]},  lanes 16-31: {B[K+2][n], B[K+3][n]}
// Since B[k][n] = W2[n][k] row-major, both fragments are contiguous float2s.
// ---------------------------------------------------------------------------
__global__ void gemm_wmma_f32_kernel(const float* __restrict__ x,
                                     const float* __restrict__ w2,
                                     float* __restrict__ h) {
    const int lane = threadIdx.x;          // 0..31
    const int wave = threadIdx.y;          // 0..7
    const long tile = (long)blockIdx.x * 8 + wave;

    const long tm = tile >> 3;             // M-tile (0..6249), 8 N-tiles
    const int  tn = (int)(tile & 7);       // N-tile (0..7)

    const int m0 = (int)tm * 16;
    const int n0 = tn * 16;
    const int half = lane >> 4;            // 0: K+0..1, 1: K+2..3
    const int l15  = lane & 15;

    const float* arow = x  + (long)(m0 + l15) * IN_C + 2 * half;
    const float* brow = w2 + (long)(n0 + l15) * IN_C + 2 * half;

    v8f acc = {};
#pragma unroll 4
    for (int k = 0; k < IN_C; k += 4) {
        v2f a = *(const v2f*)(arow + k);
        v2f b = *(const v2f*)(brow + k);
        // 8 args: (neg_a, A, neg_b, B, c_mod, C, reuse_a, reuse_b)
        acc = __builtin_amdgcn_wmma_f32_16x16x4_f32(
            false, a, false, b, (short)0, acc, false, false);
    }

    // C/D layout: VGPR r, lanes 0-15 -> h[m0+r][n0+lane];
    //                      lanes 16-31 -> h[m0+8+r][n0+lane-16]
    float* hp = h + (long)(m0 + 8 * half) * OUT_C + n0 + l15;
#pragma unroll
    for (int r = 0; r < 8; ++r) {
        hp[(long)r * OUT_C] = acc[r];
    }
}

// ---------------------------------------------------------------------------
// Message passing scatter: out[col[e], :] += h[row[e], :] * ew[e]
// One lane = (edge, 4 channels); one wave32 = one full edge (32*4 = 128 ch).
// h and out both fit in the 192 MB L2, so gathers + atomics resolve in L2.
// ---------------------------------------------------------------------------
__global__ void scatter_kernel(const int* __restrict__ row,
                               const int* __restrict__ col,
                               const float* __restrict__ ew,
                               const float* __restrict__ h,
                               float* __restrict__ out) {
    long idx = (long)blockIdx.x * blockDim.x + threadIdx.x;
    long e = idx >> 5;                     // 32 lanes per edge
    if (e >= (long)N_EDGES) return;
    int c4 = (int)(idx & 31);              // channel group (4 floats)

    int r  = row[e];
    int cl = col[e];
    float wgt = ew[e];

    const float4 v = *(const float4*)(h + (long)r * OUT_C + c4 * 4);
    float* op = out + (long)cl * OUT_C + c4 * 4;
    atomicAdd(op + 0, v.x * wgt);
    atomicAdd(op + 1, v.y * wgt);
    atomicAdd(op + 2, v.z * wgt);
    atomicAdd(op + 3, v.w * wgt);
}

// ---------------------------------------------------------------------------
// Launch
// ---------------------------------------------------------------------------
extern "C" void kernel_launch(void* const* d_in, const int* in_sizes, int n_in,
                              void* d_out, int out_size, void* d_ws, size_t ws_size,
                              hipStream_t stream) {
    const float* x  = (const float*)d_in[0];                 // [N, 256]
    const int*   ei = (const int*)d_in[1];                   // [2, E] (int32)
    const float* w  = (const float*)d_in[2];                 // [128, 256]
    const int* row = ei;                                     // sources
    const int* col = ei + N_EDGES;                           // targets
    float* out = (float*)d_out;                              // [N, 128]

    // Workspace carve-up (256B aligned slices)
    auto align256 = [](size_t v) { return (v + 255) & ~(size_t)255; };
    char* ws = (char*)d_ws;
    size_t off = 0;
    float* deg = (float*)(ws + off); off += align256((size_t)N_NODES * 4);
    float* ew  = (float*)(ws + off); off += align256((size_t)N_EDGES * 4);
    float* w2  = (float*)(ws + off); off += align256((size_t)OUT_C * IN_C * 4);
    float* h   = (float*)(ws + off); off += align256((size_t)N_NODES * OUT_C * 4);
    (void)off; (void)ws_size; (void)in_sizes; (void)n_in; (void)out_size;

    // 1) zero deg and out (scatter accumulates into them)
    zero_f32_kernel<<<2048, 256, 0, stream>>>(deg, (long)N_NODES);
    zero_f32_kernel<<<8192, 256, 0, stream>>>(out, (long)N_NODES * OUT_C);

    // 2) in-degree
    degree_kernel<<<(N_EDGES + 255) / 256, 256, 0, stream>>>(col, deg);

    // 3) per-edge normalization weights
    edge_weight_kernel<<<(N_EDGES + 255) / 256, 256, 0, stream>>>(row, col, deg, ew);

    // 4) fold postprocess into weight
    fold_weight_kernel<<<(OUT_C * IN_C + 255) / 256, 256, 0, stream>>>(w, w2);

    // 5) WMMA GEMM: 6250 M-tiles x 8 N-tiles = 50000 waves = 6250 blocks of 8
    gemm_wmma_f32_kernel<<<6250, dim3(32, 8), 0, stream>>>(x, w2, h);

    // 6) gather-scale-scatter: E*32 lanes, 256/block
    long scatter_threads = (long)N_EDGES * 32;
    int scatter_blocks = (int)((scatter_threads + 255) / 256);
    scatter_kernel<<<scatter_blocks, 256, 0, stream>>>(row, col, ew, h, out);
}